// MultiScaleSSMLayer_62302795596480
// MI455X (gfx1250) — compile-verified
//
#include <hip/hip_runtime.h>
#include <math.h>

// ---------------------------------------------------------------------------
// MultiScaleSSMLayer for MI455X (gfx1250, wave32)
// B=8, T=2048, D=1024, N_STATE=512, K=8, N_PER=64
// ---------------------------------------------------------------------------

typedef float v2f __attribute__((ext_vector_type(2)));
typedef float v8f __attribute__((ext_vector_type(8)));

#define WMMA_F32(a, b, c) \
  __builtin_amdgcn_wmma_f32_16x16x4_f32(false, (a), false, (b), (short)0, (c), false, false)

static constexpr int BATCH   = 8;
static constexpr int TLEN    = 2048;
static constexpr int DMODEL  = 1024;
static constexpr int NSTATE  = 512;
static constexpr int MTOK    = BATCH * TLEN;   // 16384 token rows

// dt[k] = 1e-3 * (1000)^(k/7), matches DT_SCALES
__device__ __constant__ float c_dts[8] = {
  0.001f, 0.0026826958f, 0.0071968567f, 0.0193069773f,
  0.0517947468f, 0.1389495494f, 0.3727593720f, 1.0f
};

__device__ __forceinline__ float softplus_f(float z) {
  return (z > 20.f) ? z : log1pf(expf(z));
}

// ---------------------------------------------------------------------------
// Kernel 1: per-token small projections -> C matrix (8x8) and alpha (softmax)
// One block per token; x row staged in LDS. 72 logit columns:
// [0..7]=diag, [8..63]=off (56), [64..71]=alpha.
// ---------------------------------------------------------------------------
__global__ __launch_bounds__(128)
void proj_small_kernel(const float* __restrict__ x,
                       const float* __restrict__ Wdiag, const float* __restrict__ bdiag,
                       const float* __restrict__ Woff,  const float* __restrict__ boff,
                       const float* __restrict__ Walpha,const float* __restrict__ balpha,
                       float* __restrict__ Cmat,        // (MTOK, 64)
                       float* __restrict__ alphaOut)    // (MTOK, 8)
{
  __shared__ float xs[DMODEL];
  __shared__ float logits[72];

  const size_t tok = blockIdx.x;
  const int tid = threadIdx.x;
  const float* xr = x + tok * DMODEL;

  for (int i = tid; i < DMODEL; i += 128) xs[i] = xr[i];
  __syncthreads();

  if (tid < 72) {
    const float* W; int stride; float bias;
    if (tid < 8)       { W = Wdiag  + tid;        stride = 8;  bias = bdiag[tid]; }
    else if (tid < 64) { W = Woff   + (tid - 8);  stride = 56; bias = boff[tid - 8]; }
    else               { W = Walpha + (tid - 64); stride = 8;  bias = balpha[tid - 64]; }
    float s = bias;
    for (int i = 0; i < DMODEL; ++i) s += xs[i] * W[(size_t)i * stride];
    logits[tid] = s;
  }
  __syncthreads();

  if (tid < 64) {
    const int i = tid >> 3, j = tid & 7;
    float val;
    if (i == j) {
      val = 1.f / (1.f + expf(-logits[i]));           // sigmoid(diag)
    } else {
      const int jp = (j < i) ? j : j - 1;             // row-major (i,j), i!=j
      const int p  = i * 7 + jp;
      val = -softplus_f(logits[8 + p]);
    }
    Cmat[tok * 64 + tid] = val;
  }
  if (tid == 64) {
    float m = logits[64];
    #pragma unroll
    for (int k = 1; k < 8; ++k) m = fmaxf(m, logits[64 + k]);
    float e[8]; float s = 0.f;
    #pragma unroll
    for (int k = 0; k < 8; ++k) { e[k] = expf(logits[64 + k] - m); s += e[k]; }
    const float inv = 1.f / s;
    #pragma unroll
    for (int k = 0; k < 8; ++k) alphaOut[tok * 8 + k] = e[k] * inv;
  }
}

// ---------------------------------------------------------------------------
// fp32 WMMA GEMM core: one wave computes a 32x64 tile of D = A(MxKd)*B(KdxN).
// N and Kd are compile-time so all addressing is shifts/immediates, and the
// B panel pointer is strictly incremented (no per-iteration multiply).
// acc[2][4] is left in registers; epilogues differ per caller.
// ---------------------------------------------------------------------------
template <int N, int Kd>
__device__ __forceinline__ void wmma_tile_32x64(const float* __restrict__ A,
                                                const float* __restrict__ Bm,
                                                int m0, int n0,
                                                int half, int l16,
                                                v8f acc[2][4])
{
  #pragma unroll
  for (int mt = 0; mt < 2; ++mt)
    #pragma unroll
    for (int nt = 0; nt < 4; ++nt)
      acc[mt][nt] = (v8f)0.f;

  const float* aBase0 = A + (size_t)(m0 + l16)      * Kd + 2 * half;
  const float* aBase1 = A + (size_t)(m0 + 16 + l16) * Kd + 2 * half;
  const float* bp     = Bm + (size_t)(2 * half) * N + n0 + l16;

  #pragma unroll 4
  for (int kk = 0; kk < Kd; kk += 4) {
    v2f a0 = *(const v2f*)(aBase0 + kk);
    v2f a1 = *(const v2f*)(aBase1 + kk);
    #pragma unroll
    for (int nt = 0; nt < 4; ++nt) {
      v2f b; b.x = bp[nt * 16]; b.y = bp[nt * 16 + N];
      acc[0][nt] = WMMA_F32(a0, b, acc[0][nt]);
      acc[1][nt] = WMMA_F32(a1, b, acc[1][nt]);
    }
    bp += 4 * N;
  }
}

// ---------------------------------------------------------------------------
// Kernel 2: Bx = x @ Wb + bb[n]
// ---------------------------------------------------------------------------
template <int N, int Kd>
__global__ __launch_bounds__(32)
void gemm_bias_n_kernel(const float* __restrict__ A, const float* __restrict__ Bm,
                        const float* __restrict__ bias, float* __restrict__ D)
{
  const int lane = threadIdx.x & 31;
  const int half = lane >> 4;
  const int l16  = lane & 15;
  const int m0 = (blockIdx.x / (N / 64)) * 32;
  const int n0 = (blockIdx.x % (N / 64)) * 64;

  v8f acc[2][4];
  wmma_tile_32x64<N, Kd>(A, Bm, m0, n0, half, l16, acc);

  #pragma unroll
  for (int mt = 0; mt < 2; ++mt)
    #pragma unroll
    for (int nt = 0; nt < 4; ++nt) {
      const int col = n0 + nt * 16 + l16;
      const float bv = bias[col];
      #pragma unroll
      for (int r = 0; r < 8; ++r) {
        const int row = m0 + mt * 16 + r + 8 * half;
        D[(size_t)row * N + col] = acc[mt][nt][r] + bv;
      }
    }
}

// ---------------------------------------------------------------------------
// Kernel 3: sequential scan over T. One block per batch, 512 threads
// (thread n owns state element n). V^T h via wave32 shuffle reduce + LDS.
// Emits g[tok,n] = alpha[tok, n>>6] * h[tok,n] for the output GEMM.
// ---------------------------------------------------------------------------
__global__ __launch_bounds__(512)
void scan_kernel(const float* __restrict__ Bx,     // (MTOK, 512)
                 const float* __restrict__ Cmat,   // (MTOK, 64)
                 const float* __restrict__ alpha,  // (MTOK, 8)
                 const float* __restrict__ U,      // (512, 8)
                 const float* __restrict__ V,      // (512, 8)
                 const float* __restrict__ log_lambda, // (8,64) flat = 512
                 float* __restrict__ g)            // (MTOK, 512)
{
  const int b    = blockIdx.x;
  const int n    = threadIdx.x;   // 0..511
  const int kblk = n >> 6;
  const int wave = n >> 5;        // 0..15
  const int lane = n & 31;

  __shared__ float partial[16][8];
  __shared__ float vhS[8];
  __shared__ float wS[8];

  float Vreg[8], Ureg[8];
  #pragma unroll
  for (int k = 0; k < 8; ++k) { Vreg[k] = V[n * 8 + k]; Ureg[k] = U[n * 8 + k]; }

  const float Adiag = expf(-c_dts[kblk] * softplus_f(log_lambda[n]));

  float h = 0.f;
  for (int t = 0; t < TLEN; ++t) {
    const size_t tok = (size_t)b * TLEN + t;

    // Vh[k] = sum_n h[n] * V[n][k] : wave shuffle reduce, then LDS combine
    #pragma unroll
    for (int k = 0; k < 8; ++k) {
      float v = h * Vreg[k];
      v += __shfl_xor(v, 1, 32);
      v += __shfl_xor(v, 2, 32);
      v += __shfl_xor(v, 4, 32);
      v += __shfl_xor(v, 8, 32);
      v += __shfl_xor(v, 16, 32);
      if (lane == 0) partial[wave][k] = v;
    }
    __syncthreads();
    if (n < 8) {
      float s = 0.f;
      #pragma unroll
      for (int w = 0; w < 16; ++w) s += partial[w][n];
      vhS[n] = s;
    }
    __syncthreads();
    if (n < 8) {                           // w[k] = C_t[k,:] . Vh
      const float* Crow = Cmat + tok * 64 + n * 8;
      float s = 0.f;
      #pragma unroll
      for (int j = 0; j < 8; ++j) s += Crow[j] * vhS[j];
      wS[n] = s;
    }
    __syncthreads();

    float coup = 0.f;
    #pragma unroll
    for (int k = 0; k < 8; ++k) coup += Ureg[k] * wS[k];

    h = Adiag * h + coup + Bx[tok * NSTATE + n];
    g[tok * NSTATE + n] = alpha[tok * 8 + kblk] * h;
    __syncthreads();   // keep partial/vhS/wS safe for next iteration
  }
}

// ---------------------------------------------------------------------------
// Kernel 4: y = g(Mx512) @ Wc(512x1024) + sum_k alpha[row,k]*bc[k,col]
// ---------------------------------------------------------------------------
template <int N, int Kd>
__global__ __launch_bounds__(32)
void gemm_out_kernel(const float* __restrict__ A,      // g
                     const float* __restrict__ Bm,     // Wc flat (512,1024)
                     const float* __restrict__ alpha,  // (M,8)
                     const float* __restrict__ bc,     // (8,1024)
                     float* __restrict__ D)
{
  const int lane = threadIdx.x & 31;
  const int half = lane >> 4;
  const int l16  = lane & 15;
  const int m0 = (blockIdx.x / (N / 64)) * 32;
  const int n0 = (blockIdx.x % (N / 64)) * 64;

  v8f acc[2][4];
  wmma_tile_32x64<N, Kd>(A, Bm, m0, n0, half, l16, acc);

  #pragma unroll
  for (int mt = 0; mt < 2; ++mt) {
    #pragma unroll
    for (int r = 0; r < 8; ++r) {
      const int row = m0 + mt * 16 + r + 8 * half;
      const float* al = alpha + (size_t)row * 8;
      float a8[8];
      #pragma unroll
      for (int k = 0; k < 8; ++k) a8[k] = al[k];
      #pragma unroll
      for (int nt = 0; nt < 4; ++nt) {
        const int col = n0 + nt * 16 + l16;
        float s = acc[mt][nt][r];
        #pragma unroll
        for (int k = 0; k < 8; ++k) s += a8[k] * bc[(size_t)k * N + col];
        D[(size_t)row * N + col] = s;
      }
    }
  }
}

// ---------------------------------------------------------------------------
// Host entry
// ---------------------------------------------------------------------------
extern "C" void kernel_launch(void* const* d_in, const int* in_sizes, int n_in,
                              void* d_out, int out_size, void* d_ws, size_t ws_size,
                              hipStream_t stream)
{
  const float* x          = (const float*)d_in[0];
  const float* Wb         = (const float*)d_in[1];
  const float* bb         = (const float*)d_in[2];
  const float* Wdiag      = (const float*)d_in[3];
  const float* bdiag      = (const float*)d_in[4];
  const float* Woff       = (const float*)d_in[5];
  const float* boff       = (const float*)d_in[6];
  const float* Walpha     = (const float*)d_in[7];
  const float* balpha     = (const float*)d_in[8];
  const float* Wc         = (const float*)d_in[9];    // (8,64,1024) == flat (512,1024)
  const float* bc         = (const float*)d_in[10];   // (8,1024)
  const float* log_lambda = (const float*)d_in[11];   // (8,64) flat 512
  const float* U          = (const float*)d_in[12];   // (512,8)
  const float* V          = (const float*)d_in[13];   // (512,8)
  float* out = (float*)d_out;

  float* ws    = (float*)d_ws;
  float* Bx    = ws;                                   // MTOK*512
  float* g     = Bx   + (size_t)MTOK * NSTATE;         // MTOK*512
  float* Cmat  = g    + (size_t)MTOK * NSTATE;         // MTOK*64
  float* alpha = Cmat + (size_t)MTOK * 64;             // MTOK*8

  // 1) small projections -> C matrices + alpha
  proj_small_kernel<<<MTOK, 128, 0, stream>>>(x, Wdiag, bdiag, Woff, boff,
                                              Walpha, balpha, Cmat, alpha);

  // 2) Bx = x @ Wb + bb   (WMMA fp32)
  gemm_bias_n_kernel<NSTATE, DMODEL>
      <<<(MTOK / 32) * (NSTATE / 64), 32, 0, stream>>>(x, Wb, bb, Bx);

  // 3) sequential scan -> g = alpha-weighted hidden states
  scan_kernel<<<BATCH, NSTATE, 0, stream>>>(Bx, Cmat, alpha, U, V, log_lambda, g);

  // 4) y = g @ Wc + alpha @ bc   (WMMA fp32)
  gemm_out_kernel<DMODEL, NSTATE>
      <<<(MTOK / 32) * (DMODEL / 64), 32, 0, stream>>>(g, Wc, alpha, bc, out);
}